// GPTGRNNDecoderS_17506286698869
// MI455X (gfx1250) — compile-verified
//
#include <hip/hip_runtime.h>
#include <math.h>

typedef float v2f __attribute__((ext_vector_type(2)));
typedef float v8f __attribute__((ext_vector_type(8)));

#define B 64
#define H 512
#define U 512
#define MAXN 512
#define M 32
#define INSZ 543           // H + M - 1
#define INSP 544           // padded K (multiple of 4), pad col is zero
#define G3H 1536           // 3*H
#define NEGV (-1e30f)

// d_out flat layout (return order: pointer, closs, enc, adj)
#define OUT_CLOSS 1048576            // B*MAXN*M
#define OUT_ENC   1048577
#define OUT_ADJ   17825793           // OUT_ENC + B*MAXN*H

// ---------------------------------------------------------------- init
__global__ void k_init(const float* __restrict__ hg, const int* __restrict__ lengths,
                       float* __restrict__ out, float* __restrict__ hid,
                       float* __restrict__ cov, float* __restrict__ dec_in,
                       int* __restrict__ offs) {
  int gid = blockIdx.x * blockDim.x + threadIdx.x;
  int nth = gridDim.x * blockDim.x;
  for (int i = gid; i < B * MAXN * M; i += nth) out[i] = 0.0f;  // pointer region
  if (gid < B * H)    hid[gid] = hg[gid];
  if (gid < B * MAXN) cov[gid] = 0.0f;
  if (gid < B * INSP) dec_in[gid] = 0.0f;   // includes pad col 543 (stays 0 forever)
  if (gid == 0) {
    int acc = 0;
    for (int b = 0; b < B; ++b) { offs[b] = acc; acc += lengths[b]; }
  }
}

// ---------------------------------------------------------------- pad w_ih to K=544 (one-time)
__global__ void k_pad_wih(const float* __restrict__ w_ih, float* __restrict__ w_ih_pad) {
  int j = blockIdx.x;                 // 0..1535
  for (int k = threadIdx.x; k < INSP; k += blockDim.x)
    w_ih_pad[(long long)j * INSP + k] = (k < INSZ) ? w_ih[(long long)j * INSZ + k] : 0.0f;
}

// ---------------------------------------------------------------- enc gather
__global__ void k_build_enc(const float* __restrict__ hsrc, const int* __restrict__ lengths,
                            const int* __restrict__ offs, float* __restrict__ out_enc) {
  int b = blockIdx.x, n = blockIdx.y;
  long long base = ((long long)(b * MAXN + n)) * H;
  if (n < lengths[b]) {
    long long src = ((long long)(offs[b] + n)) * H;
    for (int h = threadIdx.x; h < H; h += blockDim.x) out_enc[base + h] = hsrc[src + h];
  } else {
    for (int h = threadIdx.x; h < H; h += blockDim.x) out_enc[base + h] = 0.0f;
  }
}

// ---------------------------------------------------------------- WMMA helper: 16x64 tile, f32
// Computes 4 accumulators C[16 x 64] += A[16 x K] * B^T[64 x K] (B given row-major cols x K).
#define WMMA_TILE_LOOP(ACC0, ACC1, ACC2, ACC3, AROW, BROW, KLEN, KSTRIDE)            \
  for (int k0 = 0; k0 < (KLEN); k0 += 4) {                                           \
    int ka = k0 + 2 * kh;                                                            \
    v2f a;  a.x = (AROW)[ka];                  a.y = (AROW)[ka + 1];                 \
    v2f b0; b0.x = (BROW)[ka];                 b0.y = (BROW)[ka + 1];                \
    v2f b1; b1.x = (BROW)[16 * (KSTRIDE) + ka]; b1.y = (BROW)[16 * (KSTRIDE) + ka + 1]; \
    v2f b2; b2.x = (BROW)[32 * (KSTRIDE) + ka]; b2.y = (BROW)[32 * (KSTRIDE) + ka + 1]; \
    v2f b3; b3.x = (BROW)[48 * (KSTRIDE) + ka]; b3.y = (BROW)[48 * (KSTRIDE) + ka + 1]; \
    ACC0 = __builtin_amdgcn_wmma_f32_16x16x4_f32(false, a, false, b0, (short)0, ACC0, false, false); \
    ACC1 = __builtin_amdgcn_wmma_f32_16x16x4_f32(false, a, false, b1, (short)0, ACC1, false, false); \
    ACC2 = __builtin_amdgcn_wmma_f32_16x16x4_f32(false, a, false, b2, (short)0, ACC2, false, false); \
    ACC3 = __builtin_amdgcn_wmma_f32_16x16x4_f32(false, a, false, b3, (short)0, ACC3, false, false); \
  }

// ---------------------------------------------------------------- enc_proj GEMM (f32 WMMA)
// ep[r,u] = sum_k enc[r,k] * W1[u,k];  rows = 32768, cols = 512, K = 512
__global__ void k_enc_proj(const float* __restrict__ enc, const float* __restrict__ W1,
                           float* __restrict__ ep) {
  int wave = threadIdx.x >> 5;
  int lane = threadIdx.x & 31;
  int row0 = blockIdx.x * 16;
  int n0   = blockIdx.y * 256 + wave * 64;
  int m    = lane & 15;
  int kh   = lane >> 4;
  v8f acc0 = {}, acc1 = {}, acc2 = {}, acc3 = {};
  const float* arow = enc + (long long)(row0 + m) * H;
  const float* brow = W1 + (long long)(n0 + m) * H;
  WMMA_TILE_LOOP(acc0, acc1, acc2, acc3, arow, brow, H, H)
  int mrow = row0 + 8 * kh;
  int ncol = n0 + m;
  for (int r = 0; r < 8; ++r) {
    long long o = (long long)(mrow + r) * U + ncol;
    ep[o]      = acc0[r];
    ep[o + 16] = acc1[r];
    ep[o + 32] = acc2[r];
    ep[o + 48] = acc3[r];
  }
}

// ---------------------------------------------------------------- GRU gates (WMMA)
// gi[b,j] = b_ih[j] + dot(dec_in[b,:543], w_ih[j,:543])   (padded to K=544)
// gh[b,j] = b_hh[j] + dot(hid[b,:512],   w_hh[j,:512])
// rows = B = 64 (4 row-tiles), cols = 1536 (grid.y*256 + wave*64)
__global__ void k_gru_gates(const float* __restrict__ dec_in, const float* __restrict__ hid,
                            const float* __restrict__ w_ih_pad, const float* __restrict__ w_hh,
                            const float* __restrict__ b_ih, const float* __restrict__ b_hh,
                            float* __restrict__ gi, float* __restrict__ gh) {
  int wave = threadIdx.x >> 5;
  int lane = threadIdx.x & 31;
  int row0 = blockIdx.x * 16;
  int n0   = blockIdx.y * 256 + wave * 64;
  int m    = lane & 15;
  int kh   = lane >> 4;
  v8f ai0 = {}, ai1 = {}, ai2 = {}, ai3 = {};
  v8f ah0 = {}, ah1 = {}, ah2 = {}, ah3 = {};
  {
    const float* arow = dec_in + (long long)(row0 + m) * INSP;
    const float* brow = w_ih_pad + (long long)(n0 + m) * INSP;
    WMMA_TILE_LOOP(ai0, ai1, ai2, ai3, arow, brow, INSP, INSP)
  }
  {
    const float* arow = hid + (long long)(row0 + m) * H;
    const float* brow = w_hh + (long long)(n0 + m) * H;
    WMMA_TILE_LOOP(ah0, ah1, ah2, ah3, arow, brow, H, H)
  }
  int mrow = row0 + 8 * kh;
  int ncol = n0 + m;
  float bi0 = b_ih[ncol], bi1 = b_ih[ncol + 16], bi2 = b_ih[ncol + 32], bi3 = b_ih[ncol + 48];
  float bh0 = b_hh[ncol], bh1 = b_hh[ncol + 16], bh2 = b_hh[ncol + 32], bh3 = b_hh[ncol + 48];
  for (int r = 0; r < 8; ++r) {
    long long o = (long long)(mrow + r) * G3H + ncol;
    gi[o]      = ai0[r] + bi0;
    gi[o + 16] = ai1[r] + bi1;
    gi[o + 32] = ai2[r] + bi2;
    gi[o + 48] = ai3[r] + bi3;
    gh[o]      = ah0[r] + bh0;
    gh[o + 16] = ah1[r] + bh1;
    gh[o + 32] = ah2[r] + bh2;
    gh[o + 48] = ah3[r] + bh3;
  }
}

__global__ void k_gru_combine(float* __restrict__ hid, const float* __restrict__ gi,
                              const float* __restrict__ gh) {
  int b = blockIdx.x;
  for (int h = threadIdx.x; h < H; h += blockDim.x) {
    float r = 1.0f / (1.0f + expf(-(gi[b * G3H + h] + gh[b * G3H + h])));
    float z = 1.0f / (1.0f + expf(-(gi[b * G3H + H + h] + gh[b * G3H + H + h])));
    float n = tanhf(gi[b * G3H + 2 * H + h] + r * gh[b * G3H + 2 * H + h]);
    float hp = hid[b * H + h];
    hid[b * H + h] = (1.0f - z) * n + z * hp;
  }
}

// ---------------------------------------------------------------- hw2 = hid @ W2^T (WMMA)
// rows = 64, cols = 512, K = 512
__global__ void k_hw2(const float* __restrict__ hid, const float* __restrict__ W2,
                      float* __restrict__ hw2) {
  int wave = threadIdx.x >> 5;
  int lane = threadIdx.x & 31;
  int row0 = blockIdx.x * 16;
  int n0   = blockIdx.y * 256 + wave * 64;
  int m    = lane & 15;
  int kh   = lane >> 4;
  v8f acc0 = {}, acc1 = {}, acc2 = {}, acc3 = {};
  const float* arow = hid + (long long)(row0 + m) * H;
  const float* brow = W2 + (long long)(n0 + m) * H;
  WMMA_TILE_LOOP(acc0, acc1, acc2, acc3, arow, brow, H, H)
  int mrow = row0 + 8 * kh;
  int ncol = n0 + m;
  for (int r = 0; r < 8; ++r) {
    long long o = (long long)(mrow + r) * U + ncol;
    hw2[o]      = acc0[r];
    hw2[o + 16] = acc1[r];
    hw2[o + 32] = acc2[r];
    hw2[o + 48] = acc3[r];
  }
}

// ---------------------------------------------------------------- theta
__global__ void k_theta(const float* __restrict__ hid, const float* __restrict__ Wf,
                        const float* __restrict__ bf, float* __restrict__ dec_in,
                        float* __restrict__ out_adj, int step) {
  int b = blockIdx.x;
  int j = threadIdx.x;
  if (j >= M - 1) return;
  const float* hp = hid + b * H;
  const float* w  = Wf + (long long)j * H;
  float s = bf[j];
  for (int k = 0; k < H; ++k) s += hp[k] * w[k];
  float th = (j < step) ? (1.0f / (1.0f + expf(-s))) : 0.0f;
  dec_in[b * INSP + H + j] = th;
  out_adj[(b * (M - 1) + j) * M + step] = th;
}

// ---------------------------------------------------------------- attention scores
__global__ void k_scores(const float* __restrict__ ep, const float* __restrict__ hw2,
                         const float* __restrict__ cov, const float* __restrict__ wc,
                         const float* __restrict__ v, const int* __restrict__ lengths,
                         float* __restrict__ scores) {
  __shared__ float sd[128];
  int b = blockIdx.x, n = blockIdx.y;
  float cv = cov[b * MAXN + n];
  const float* er = ep + (long long)(b * MAXN + n) * U;
  const float* h2 = hw2 + b * U;
  float s = 0.0f;
  for (int u = threadIdx.x; u < U; u += blockDim.x)
    s += v[u] * tanhf(er[u] + h2[u] + cv * wc[u]);
  sd[threadIdx.x] = s; __syncthreads();
  for (int st = blockDim.x >> 1; st > 0; st >>= 1) {
    if (threadIdx.x < st) sd[threadIdx.x] += sd[threadIdx.x + st];
    __syncthreads();
  }
  if (threadIdx.x == 0)
    scores[b * MAXN + n] = (n < lengths[b]) ? sd[0] : NEGV;
}

// ---------------------------------------------------------------- softmax / gumbel / argmax / update
__global__ void k_softmax(const float* __restrict__ scores, float* __restrict__ cov,
                          const float* __restrict__ gumbel, const float* __restrict__ enc,
                          float* __restrict__ dec_in, float* __restrict__ out_ptr,
                          float* __restrict__ closs_parts, int step) {
  __shared__ float sd[MAXN];
  __shared__ int   si[MAXN];
  __shared__ float bc;
  int b = blockIdx.x;
  int n = threadIdx.x;
  float sc = scores[b * MAXN + n];
  sd[n] = sc; __syncthreads();
  for (int st = MAXN >> 1; st > 0; st >>= 1) { if (n < st) sd[n] = fmaxf(sd[n], sd[n + st]); __syncthreads(); }
  float mx = sd[0]; __syncthreads();
  float e = expf(sc - mx);
  sd[n] = e; __syncthreads();
  for (int st = MAXN >> 1; st > 0; st >>= 1) { if (n < st) sd[n] += sd[n + st]; __syncthreads(); }
  float sum = sd[0]; __syncthreads();
  float att = e / sum;
  float cv = cov[b * MAXN + n];
  sd[n] = fminf(att, cv); __syncthreads();
  for (int st = MAXN >> 1; st > 0; st >>= 1) { if (n < st) sd[n] += sd[n + st]; __syncthreads(); }
  if (n == 0) closs_parts[step * B + b] = sd[0];
  __syncthreads();
  cov[b * MAXN + n] = cv + att;
  float ly = logf(att + 1e-12f) + gumbel[((long long)step * B + b) * MAXN + n];  // TAU = 1
  sd[n] = ly; __syncthreads();
  for (int st = MAXN >> 1; st > 0; st >>= 1) { if (n < st) sd[n] = fmaxf(sd[n], sd[n + st]); __syncthreads(); }
  float my = sd[0]; __syncthreads();
  float ey = expf(ly - my);
  sd[n] = ey; __syncthreads();
  for (int st = MAXN >> 1; st > 0; st >>= 1) { if (n < st) sd[n] += sd[n + st]; __syncthreads(); }
  float sy = sd[0]; __syncthreads();
  float y = ey / sy;
  sd[n] = y; si[n] = n; __syncthreads();
  for (int st = MAXN >> 1; st > 0; st >>= 1) {
    if (n < st) {
      float v1 = sd[n], v2 = sd[n + st];
      int i1 = si[n], i2 = si[n + st];
      if (v2 > v1 || (v2 == v1 && i2 < i1)) { sd[n] = v2; si[n] = i2; }
    }
    __syncthreads();
  }
  int nstar = si[0]; __syncthreads();
  if (n == nstar) bc = y;
  __syncthreads();
  float ystar = bc;
  float c = (1.0f + ystar) - ystar;   // att_hard argmax value; others are exact +0
  if (n == 0) out_ptr[((long long)(b * MAXN + nstar)) * M + step] = c;
  const float* er = enc + ((long long)(b * MAXN + nstar)) * H;
  dec_in[b * INSP + n] = er[n] * c;   // chosen gather
}

// ---------------------------------------------------------------- closs finalize (fixed order)
__global__ void k_finalize(const float* __restrict__ closs_parts, float* __restrict__ out_closs) {
  if (blockIdx.x == 0 && threadIdx.x == 0) {
    float s = 0.0f;
    for (int i = 0; i < M * B; ++i) s += closs_parts[i];
    out_closs[0] = s / (float)B;
  }
}

// ---------------------------------------------------------------- launch
extern "C" void kernel_launch(void* const* d_in, const int* in_sizes, int n_in,
                              void* d_out, int out_size, void* d_ws, size_t ws_size,
                              hipStream_t stream) {
  const float* h_in    = (const float*)d_in[0];
  const float* hg      = (const float*)d_in[1];
  const int*   lengths = (const int*)  d_in[2];
  const float* gumbel  = (const float*)d_in[3];
  const float* W1      = (const float*)d_in[4];
  const float* W2      = (const float*)d_in[5];
  const float* wc      = (const float*)d_in[6];
  const float* v       = (const float*)d_in[7];
  const float* w_ih    = (const float*)d_in[8];
  const float* w_hh    = (const float*)d_in[9];
  const float* b_ih    = (const float*)d_in[10];
  const float* b_hh    = (const float*)d_in[11];
  const float* Wf      = (const float*)d_in[12];
  const float* bf      = (const float*)d_in[13];

  float* out       = (float*)d_out;
  float* out_ptr   = out;
  float* out_closs = out + OUT_CLOSS;
  float* out_enc   = out + OUT_ENC;
  float* out_adj   = out + OUT_ADJ;

  float* ws = (float*)d_ws;
  size_t o = 0;
  float* ep          = ws + o; o += (size_t)B * MAXN * U;   // 16,777,216
  float* w_ih_pad    = ws + o; o += (size_t)G3H * INSP;     //    835,584
  float* hid         = ws + o; o += B * H;
  float* cov         = ws + o; o += B * MAXN;
  float* hw2         = ws + o; o += B * U;
  float* dec_in      = ws + o; o += B * INSP;
  float* gi          = ws + o; o += B * G3H;
  float* gh          = ws + o; o += B * G3H;
  float* scores      = ws + o; o += B * MAXN;
  float* closs_parts = ws + o; o += M * B;
  int*   offs        = (int*)(ws + o);                      // ~71.9 MB total

  k_init<<<dim3(2048), dim3(256), 0, stream>>>(hg, lengths, out, hid, cov, dec_in, offs);
  k_pad_wih<<<dim3(G3H), dim3(256), 0, stream>>>(w_ih, w_ih_pad);
  k_build_enc<<<dim3(B, MAXN), dim3(128), 0, stream>>>(h_in, lengths, offs, out_enc);
  k_enc_proj<<<dim3((B * MAXN) / 16, 2), dim3(128), 0, stream>>>(out_enc, W1, ep);

  for (int step = 0; step < M; ++step) {
    k_gru_gates<<<dim3(B / 16, G3H / 256), dim3(128), 0, stream>>>(dec_in, hid, w_ih_pad, w_hh,
                                                                   b_ih, b_hh, gi, gh);
    k_gru_combine<<<dim3(B), dim3(128), 0, stream>>>(hid, gi, gh);
    k_hw2<<<dim3(B / 16, U / 256), dim3(128), 0, stream>>>(hid, W2, hw2);
    k_theta<<<dim3(B), dim3(32), 0, stream>>>(hid, Wf, bf, dec_in, out_adj, step);
    k_scores<<<dim3(B, MAXN), dim3(128), 0, stream>>>(ep, hw2, cov, wc, v, lengths, scores);
    k_softmax<<<dim3(B), dim3(MAXN), 0, stream>>>(scores, cov, gumbel, out_enc, dec_in, out_ptr,
                                                  closs_parts, step);
  }
  k_finalize<<<1, 1, 0, stream>>>(closs_parts, out_closs);
}